// Cluster_34935263985866
// MI455X (gfx1250) — compile-verified
//
#include <hip/hip_runtime.h>

// k-means (Lloyd) for MI455X / gfx1250, wave32.
// One workgroup per batch; 25 iterations fully on-chip (centers in LDS,
// points L2-resident).  Distance GEMM uses V_WMMA_F32_16X16X4_F32 with the
// B-operand staged into LDS by the Tensor Data Mover (tensor_load_to_lds,
// double-buffered per wave, synchronized with s_wait_tensorcnt).

typedef __attribute__((ext_vector_type(2))) float        v2f;
typedef __attribute__((ext_vector_type(8))) float        v8f;
typedef __attribute__((ext_vector_type(4))) unsigned int u32x4;
typedef __attribute__((ext_vector_type(8))) unsigned int u32x8;

#define NUM_PARTS 6
#define N_ITERS   25
#define C_DIM     2048            // point dimensionality (slow axis in memory)
#define N_PTS     288             // points per batch (fast axis in memory)
#define THREADS   512             // 16 wave32 waves
#define NWAVES    (THREADS / 32)
#define N_TILES   (N_PTS / 16)    // 18 point-column tiles
#define CHUNK     32              // C-rows staged per TDM transfer (16x32 tile)
#define NCHUNKS   (C_DIM / CHUNK)

// Issue one TDM transfer: a [CHUNK rows x 16 cols] f32 tile of the [C_DIM x
// N_PTS] tensor (row stride N_PTS) -> LDS at lds_addr.  All descriptor words
// are wave-uniform; readfirstlane pins them to SGPRs for the "s" constraints.
__device__ __forceinline__ void tdm_load_tile(unsigned lds_addr,
                                              unsigned long long ga)
{
    u32x4 g0;
    g0.x = 1u;                                             // count=1, user mode
    g0.y = (unsigned)__builtin_amdgcn_readfirstlane((int)lds_addr);
    g0.z = (unsigned)__builtin_amdgcn_readfirstlane((int)(unsigned)ga);
    g0.w = (unsigned)__builtin_amdgcn_readfirstlane(
               (int)((((unsigned)(ga >> 32)) & 0x01FFFFFFu) | (2u << 30)));
    u32x8 g1;
    g1[0] = 2u << 16;                 // data_size = 4 bytes; no multicast/pad
    g1[1] = ((unsigned)N_PTS) << 16;  // tensor_dim0 = 288  (lo16 @ bits 63:48)
    g1[2] = ((unsigned)C_DIM) << 16;  // tensor_dim1 = 2048 (lo16 @ bits 95:80)
    g1[3] = 16u << 16;                // tile_dim0   = 16   (bits 127:112)
    g1[4] = (unsigned)CHUNK;          // tile_dim1   = 32, tile_dim2 = 0
    g1[5] = (unsigned)N_PTS;          // tensor_dim0_stride = 288 (lo32)
    g1[6] = 0u;
    g1[7] = 0u;
    asm volatile("tensor_load_to_lds %0, %1" :: "s"(g0), "s"(g1) : "memory");
}

__global__ __launch_bounds__(THREADS)
void kmeans_cdna5_kernel(const float* __restrict__ xg, float* __restrict__ out)
{
    __shared__ float Cc[NUM_PARTS * C_DIM];                        // 48 KB
    __shared__ __align__(128) float stage[NWAVES][2][CHUNK * 16];  // 64 KB
    __shared__ float csq[8];
    __shared__ int   cnt[8];
    __shared__ __align__(16) int asg[N_PTS];

    const int b    = blockIdx.x;
    const int tid  = threadIdx.x;
    const int lane = tid & 31;
    const int wave = tid >> 5;
    const float* __restrict__ x = xg + (size_t)b * C_DIM * N_PTS;

    // ---- init: centers0 = first 6 points ----
    for (int c = tid; c < C_DIM; c += THREADS) {
        #pragma unroll
        for (int k = 0; k < NUM_PARTS; ++k)
            Cc[k * C_DIM + c] = x[(size_t)c * N_PTS + k];   // pts[b, k, c]
    }
    __syncthreads();

    // WMMA f32 16x16x4 operand addressing:
    //   lanes 0..15  : M = lane,    VGPR0/1 = K0,K1
    //   lanes 16..31 : M = lane-16, VGPR0/1 = K2,K3
    // Rows m>=6 alias center row 0: their D rows (6..15) are never read by
    // the argmin (lanes 0..15 only inspect acc[0..5]), so no zero-padding.
    const int m    = lane & 15;
    const int kb   = (lane >> 4) * 2;
    const int mrow = (m < NUM_PARTS) ? m : 0;
    const float* __restrict__ ap = &Cc[mrow * C_DIM + kb];

    float* sb0 = &stage[wave][0][0];
    float* sb1 = &stage[wave][1][0];
    // Low 32 bits of the generic address == wave-relative LDS byte offset.
    const unsigned lds0 = (unsigned)(size_t)sb0;
    const unsigned lds1 = (unsigned)(size_t)sb1;

    for (int it = 0; it < N_ITERS; ++it) {
        // ---- zero per-iteration scalars ----
        if (tid < 8) { csq[tid] = 0.0f; cnt[tid] = 0; }
        __syncthreads();

        // ---- c_sq[k] = sum_c centers[k][c]^2 ----
        {
            float p0 = 0.f, p1 = 0.f, p2 = 0.f, p3 = 0.f, p4 = 0.f, p5 = 0.f;
            for (int c = tid; c < C_DIM; c += THREADS) {
                float t0 = Cc[0 * C_DIM + c], t1 = Cc[1 * C_DIM + c];
                float t2 = Cc[2 * C_DIM + c], t3 = Cc[3 * C_DIM + c];
                float t4 = Cc[4 * C_DIM + c], t5 = Cc[5 * C_DIM + c];
                p0 += t0 * t0; p1 += t1 * t1; p2 += t2 * t2;
                p3 += t3 * t3; p4 += t4 * t4; p5 += t5 * t5;
            }
            atomicAdd(&csq[0], p0); atomicAdd(&csq[1], p1);
            atomicAdd(&csq[2], p2); atomicAdd(&csq[3], p3);
            atomicAdd(&csq[4], p4); atomicAdd(&csq[5], p5);
        }
        __syncthreads();

        // ---- distances via WMMA, B staged by TDM (double buffered) ----
        // Dots[k][n] = sum_c Cc[k][c] * x[c*288+n]; argmin_k(csq[k]-2*Dots).
        for (int tile = wave; tile < N_TILES; tile += NWAVES) {
            const unsigned long long ga =
                (unsigned long long)(size_t)(x + tile * 16);
            tdm_load_tile(lds0, ga);                       // chunk 0 -> buf0
            v8f acc = {};
            for (int ci = 0; ci < NCHUNKS; ++ci) {
                if (ci + 1 < NCHUNKS) {                    // prefetch next
                    tdm_load_tile((ci & 1) ? lds0 : lds1,
                                  ga + (unsigned long long)(ci + 1) *
                                       (CHUNK * N_PTS * 4ull));
                    __builtin_amdgcn_s_wait_tensorcnt(1);  // chunk ci ready
                } else {
                    __builtin_amdgcn_s_wait_tensorcnt(0);
                }
                const float* __restrict__ sb  = (ci & 1) ? sb1 : sb0;
                const float* __restrict__ apc = ap + ci * CHUNK;
                #pragma unroll
                for (int cl = 0; cl < CHUNK; cl += 4) {
                    v2f a = *(const v2f*)(apc + cl);       // ds_load_b64
                    v2f bv;
                    bv.x = sb[(cl + kb)     * 16 + m];     // staged x[c][n]
                    bv.y = sb[(cl + kb + 1) * 16 + m];
                    acc = __builtin_amdgcn_wmma_f32_16x16x4_f32(
                        false, a, false, bv, (short)0, acc, false, false);
                }
            }
            if (lane < 16) {
                const int n = tile * 16 + m;
                float best = csq[0] - 2.0f * acc[0]; int bk = 0;
                #pragma unroll
                for (int r = 1; r < NUM_PARTS; ++r) {
                    float d = csq[r] - 2.0f * acc[r];
                    if (d < best) { best = d; bk = r; }
                }
                asg[n] = bk;
            }
        }
        __syncthreads();

        // ---- counts ----
        for (int n = tid; n < N_PTS; n += THREADS) atomicAdd(&cnt[asg[n]], 1);
        __syncthreads();

        // ---- centroid sums + update (assign[n] is wave-uniform here) ----
        for (int c = tid; c < C_DIM; c += THREADS) {
            float a0 = 0.f, a1 = 0.f, a2 = 0.f, a3 = 0.f, a4 = 0.f, a5 = 0.f;
            const float4* __restrict__ row = (const float4*)(x + (size_t)c * N_PTS);
            const int4*   __restrict__ ks4 = (const int4*)asg;
            #pragma unroll 2
            for (int n4 = 0; n4 < N_PTS / 4; ++n4) {
                float4 v = row[n4];
                int4   k = ks4[n4];
#define ACCUM(KK, VV) do {                                            \
                int ku_ = __builtin_amdgcn_readfirstlane(KK);         \
                float v_ = (VV);                                      \
                if      (ku_ == 0) a0 += v_;                          \
                else if (ku_ == 1) a1 += v_;                          \
                else if (ku_ == 2) a2 += v_;                          \
                else if (ku_ == 3) a3 += v_;                          \
                else if (ku_ == 4) a4 += v_;                          \
                else               a5 += v_; } while (0)
                ACCUM(k.x, v.x); ACCUM(k.y, v.y);
                ACCUM(k.z, v.z); ACCUM(k.w, v.w);
#undef ACCUM
            }
            Cc[0 * C_DIM + c] = (cnt[0] > 0) ? a0 / (float)cnt[0] : Cc[0 * C_DIM + c];
            Cc[1 * C_DIM + c] = (cnt[1] > 0) ? a1 / (float)cnt[1] : Cc[1 * C_DIM + c];
            Cc[2 * C_DIM + c] = (cnt[2] > 0) ? a2 / (float)cnt[2] : Cc[2 * C_DIM + c];
            Cc[3 * C_DIM + c] = (cnt[3] > 0) ? a3 / (float)cnt[3] : Cc[3 * C_DIM + c];
            Cc[4 * C_DIM + c] = (cnt[4] > 0) ? a4 / (float)cnt[4] : Cc[4 * C_DIM + c];
            Cc[5 * C_DIM + c] = (cnt[5] > 0) ? a5 / (float)cnt[5] : Cc[5 * C_DIM + c];
        }
        __syncthreads();
    }

    // ---- output: clus[b][c][k] = centers[k][c] ----
    for (int c = tid; c < C_DIM; c += THREADS) {
        float* __restrict__ o = out + ((size_t)b * C_DIM + c) * NUM_PARTS;
        #pragma unroll
        for (int k = 0; k < NUM_PARTS; ++k) o[k] = Cc[k * C_DIM + c];
    }
}

extern "C" void kernel_launch(void* const* d_in, const int* in_sizes, int n_in,
                              void* d_out, int out_size, void* d_ws, size_t ws_size,
                              hipStream_t stream)
{
    (void)n_in; (void)out_size; (void)d_ws; (void)ws_size;
    const float* x   = (const float*)d_in[0];
    float*       out = (float*)d_out;
    const int B = in_sizes[0] / (C_DIM * N_PTS);   // 64
    kmeans_cdna5_kernel<<<dim3(B), dim3(THREADS), 0, stream>>>(x, out);
}